// CPCCLoss_62886911148250
// MI455X (gfx1250) — compile-verified
//
#include <hip/hip_runtime.h>
#include <hip/hip_bf16.h>

// Problem constants (match reference)
#define PB   128     // prototype groups P
#define BB   4096    // batch B
#define DD   128     // feature dim D
#define CPC_EPS 1e-6f
#define MAXNORM 0.99999f   // (1 - 1e-5)/sqrt(c), c = 1
#define ITERS 10

typedef float v2f __attribute__((ext_vector_type(2)));
typedef float v8f __attribute__((ext_vector_type(8)));

// ---------------------------------------------------------------------------
// Block-wide sum reduction (blockDim.x threads, power of 2), shared buffer sm
// ---------------------------------------------------------------------------
__device__ __forceinline__ float block_reduce_sum(float v, float* sm) {
    int t = threadIdx.x;
    sm[t] = v;
    __syncthreads();
    for (int s = blockDim.x >> 1; s > 0; s >>= 1) {
        if (t < s) sm[t] += sm[t + s];
        __syncthreads();
    }
    float r = sm[0];
    __syncthreads();
    return r;
}

// ---------------------------------------------------------------------------
// 1) expmap0: map each row of representations onto the Poincare ball.
//    One block (128 threads) per row; also emits x2_b = |xp_b|^2.
// ---------------------------------------------------------------------------
__global__ void expmap0_kernel(const float* __restrict__ reps,
                               float* __restrict__ Xp,
                               float* __restrict__ x2) {
    __shared__ float sm[DD];
    int b = blockIdx.x, d = threadIdx.x;
    float v = reps[b * DD + d];
    float n2 = block_reduce_sum(v * v, sm);
    float n = fmaxf(sqrtf(n2), CPC_EPS);
    float s = tanhf(n) / n;           // sc = sqrt(c) = 1
    float xp = v * s;
    Xp[b * DD + d] = xp;
    float xx = block_reduce_sum(xp * xp, sm);
    if (d == 0) x2[b] = xx;
}

// ---------------------------------------------------------------------------
// 2) Euclidean group-mean init + project_to_ball.
//    One block per group p; thread d owns component d.
// ---------------------------------------------------------------------------
__global__ void init_mu_kernel(const float* __restrict__ Xp,
                               const int* __restrict__ targets,
                               float* __restrict__ cnt,
                               float* __restrict__ Mu,
                               float* __restrict__ Mu2) {
    __shared__ float sm[DD];
    int p = blockIdx.x, d = threadIdx.x;
    float acc = 0.f;
    int c = 0;
    for (int b = 0; b < BB; ++b) {
        if (targets[b] == p) { acc += Xp[b * DD + d]; ++c; }
    }
    float cp = fmaxf((float)c, 1.f);
    float mu = acc / cp;
    float n2 = block_reduce_sum(mu * mu, sm);
    float n = sqrtf(n2);
    float factor = fminf(1.f, MAXNORM / fmaxf(n, CPC_EPS));
    mu *= factor;
    Mu[p * DD + d] = mu;
    if (d == 0) {
        float nn = n * factor;
        Mu2[p] = nn * nn;
        cnt[p] = cp;
    }
}

// ---------------------------------------------------------------------------
// 3) fp32 WMMA GEMM:  C[M,N] = A[M,K] * B[N,K]^T   (row-major, dims % 16 == 0)
//    One wave32 per 16x16 output tile; K stepped 4 at a time with
//    v_wmma_f32_16x16x4_f32.
//    A operand layout: lane L (M = L&15) VGPR0/1 hold K = k + 2*(L>>4) (+1).
//    B (= row of the [N,K] matrix, i.e. column of B^T) uses the same pattern
//    with N = L&15. C/D: VGPR r holds row M = r + 8*(L>>4), col N = L&15.
// ---------------------------------------------------------------------------
__global__ void gemm_abt_wmma(const float* __restrict__ A,
                              const float* __restrict__ Bm,
                              float* __restrict__ C,
                              int K, int lda, int ldb, int ldc) {
    int lane = threadIdx.x;           // 0..31, one wave
    int half = lane >> 4;             // 0 or 1
    int l16  = lane & 15;
    int mrow = blockIdx.y * 16 + l16; // A row for this lane
    int nrow = blockIdx.x * 16 + l16; // B row (column of B^T)

    v8f acc = {};
    for (int k = 0; k < K; k += 4) {
        int kk = k + half * 2;
        v2f a = *(const v2f*)(A  + mrow * lda + kk);
        v2f b = *(const v2f*)(Bm + nrow * ldb + kk);
        acc = __builtin_amdgcn_wmma_f32_16x16x4_f32(
            /*neg_a=*/false, a, /*neg_b=*/false, b,
            /*c_mod=*/(short)0, acc, /*reuse_a=*/false, /*reuse_b=*/false);
    }
    int cbase = blockIdx.y * 16 + half * 8;
    int ccol  = blockIdx.x * 16 + l16;
#pragma unroll
    for (int r = 0; r < 8; ++r) {
        C[(cbase + r) * ldc + ccol] = acc[r];
    }
}

// ---------------------------------------------------------------------------
// 4) One Karcher-style iteration of the masked Mobius mean.
//    mobius_add(x_b, -mu_p) = alpha_b * x_b - beta_b * mu_p, with
//      den   = max(1 - 2 s + x2_b*y2_p, EPS),  s = x_b . mu_p  (from S)
//      alpha = (1 - 2 s + y2_p)/den,  beta = (1 - x2_b)/den
//    Then mu <- project(mobius_add(mu, mdiff)).  One block per group.
// ---------------------------------------------------------------------------
__global__ void update_mu_kernel(const float* __restrict__ Xp,
                                 const float* __restrict__ x2,
                                 const int* __restrict__ targets,
                                 const float* __restrict__ cnt,
                                 const float* __restrict__ S,
                                 float* __restrict__ Mu,
                                 float* __restrict__ Mu2) {
    __shared__ float sm[DD];
    int p = blockIdx.x, d = threadIdx.x;
    float mu = Mu[p * DD + d];
    float y2 = Mu2[p];
    float cp = cnt[p];

    float vec = 0.f, scal = 0.f;
    for (int b = 0; b < BB; ++b) {
        if (targets[b] == p) {
            float s   = S[b * PB + p];
            float x2b = x2[b];
            float den = fmaxf(1.f - 2.f * s + x2b * y2, CPC_EPS);
            float alpha = (1.f - 2.f * s + y2) / den;
            float beta  = (1.f - x2b) / den;
            vec  += alpha * Xp[b * DD + d];
            scal += beta;
        }
    }
    float md = vec / cp - (scal / cp) * mu;   // mdiff component d

    // mobius_add(mu, mdiff)
    float xy  = block_reduce_sum(mu * md, sm);
    float y2m = block_reduce_sum(md * md, sm);
    float den2 = fmaxf(1.f + 2.f * xy + y2 * y2m, CPC_EPS);
    float t = ((1.f + 2.f * xy + y2m) * mu + (1.f - y2) * md) / den2;

    // project_to_ball
    float n2 = block_reduce_sum(t * t, sm);
    float n = sqrtf(n2);
    float factor = fminf(1.f, MAXNORM / fmaxf(n, CPC_EPS));
    float outd = t * factor;
    Mu[p * DD + d] = outd;
    if (d == 0) {
        float nn = n * factor;
        Mu2[p] = nn * nn;
    }
}

// ---------------------------------------------------------------------------
// 5) Poincare distances from the Gram matrix + deterministic single-block
//    Pearson correlation against dT over the strict upper triangle.
// ---------------------------------------------------------------------------
__global__ void cpcc_final_kernel(const float* __restrict__ Gram,
                                  const int* __restrict__ dT,
                                  float* __restrict__ out) {
    __shared__ float sm[256];
    int t = threadIdx.x;
    float sx = 0.f, sy = 0.f, sxx = 0.f, syy = 0.f, sxy = 0.f;
    for (int idx = t; idx < PB * PB; idx += 256) {
        int i = idx / PB, j = idx % PB;
        if (j <= i) continue;
        float gii = Gram[i * PB + i];
        float gjj = Gram[j * PB + j];
        float gij = Gram[i * PB + j];
        float dist2 = gii + gjj - 2.f * gij;
        float ai = fmaxf(1.f - gii, CPC_EPS);
        float aj = fmaxf(1.f - gjj, CPC_EPS);
        float z = 1.f + (2.f * dist2) / fmaxf(ai * aj, CPC_EPS);
        z = fmaxf(z, 1.f + CPC_EPS);
        float dB = logf(z + sqrtf(z * z - 1.f));   // arccosh
        int k = i * (2 * PB - i - 1) / 2 + (j - i - 1); // triu linear index
        float y = (float)dT[k];
        sx += dB; sy += y; sxx += dB * dB; syy += y * y; sxy += dB * y;
    }
    sx  = block_reduce_sum(sx,  sm);
    sy  = block_reduce_sum(sy,  sm);
    sxx = block_reduce_sum(sxx, sm);
    syy = block_reduce_sum(syy, sm);
    sxy = block_reduce_sum(sxy, sm);
    if (t == 0) {
        float n = (float)(PB * (PB - 1) / 2);
        float cov = n * sxy - sx * sy;
        float vx  = n * sxx - sx * sx;
        float vy  = n * syy - sy * sy;
        float corr = cov / sqrtf(vx * vy);
        out[0] = isnan(corr) ? 1.0f : 1.0f - corr;
    }
}

// ---------------------------------------------------------------------------
// Host orchestration (graph-capture safe: only kernel launches on `stream`).
// Workspace layout (floats): Xp | x2 | cnt | Mu | Mu2 | S | Gram  (~4.2 MB)
// ---------------------------------------------------------------------------
extern "C" void kernel_launch(void* const* d_in, const int* in_sizes, int n_in,
                              void* d_out, int out_size, void* d_ws, size_t ws_size,
                              hipStream_t stream) {
    (void)in_sizes; (void)n_in; (void)out_size; (void)ws_size;

    const float* reps    = (const float*)d_in[0];
    const int*   targets = (const int*)d_in[1];
    const int*   dT      = (const int*)d_in[2];
    float*       out     = (float*)d_out;

    float* ws   = (float*)d_ws;
    float* Xp   = ws;                       // BB*DD
    float* x2   = Xp   + (size_t)BB * DD;   // BB
    float* cnt  = x2   + BB;                // PB
    float* Mu   = cnt  + PB;                // PB*DD
    float* Mu2  = Mu   + (size_t)PB * DD;   // PB
    float* S    = Mu2  + PB;                // BB*PB
    float* Gram = S    + (size_t)BB * PB;   // PB*PB

    // Map to ball + per-point squared norms
    expmap0_kernel<<<BB, DD, 0, stream>>>(reps, Xp, x2);

    // Euclidean group mean init (projected)
    init_mu_kernel<<<PB, DD, 0, stream>>>(Xp, targets, cnt, Mu, Mu2);

    // Karcher iterations: S = Xp * Mu^T via fp32 WMMA, then Mobius update
    dim3 gS(PB / 16, BB / 16);
    for (int it = 0; it < ITERS; ++it) {
        gemm_abt_wmma<<<gS, 32, 0, stream>>>(Xp, Mu, S, DD, DD, DD, PB);
        update_mu_kernel<<<PB, DD, 0, stream>>>(Xp, x2, targets, cnt, S, Mu, Mu2);
    }

    // Prototype Gram matrix via fp32 WMMA
    dim3 gG(PB / 16, PB / 16);
    gemm_abt_wmma<<<gG, 32, 0, stream>>>(Mu, Mu, Gram, DD, DD, DD, PB);

    // Distances + Pearson correlation -> scalar loss
    cpcc_final_kernel<<<1, 256, 0, stream>>>(Gram, dT, out);
}